// Global_Reason_Unit_17824114278435
// MI455X (gfx1250) — compile-verified
//
#include <hip/hip_runtime.h>
#include <math.h>

#define BSZ    8
#define C_IN   512
#define NPIX   9216      // 96*96
#define NNODE  64
#define NSTATE 128
#define NSTRIP 576       // NPIX/16

typedef __attribute__((ext_vector_type(2))) float v2f;
typedef __attribute__((ext_vector_type(8))) float v8f;

__device__ __forceinline__ v8f wmma_f32(v2f a, v2f b, v8f c) {
    // V_WMMA_F32_16X16X4_F32 : D = A(16x4) * B(4x16) + C(16x16), all fp32
    return __builtin_amdgcn_wmma_f32_16x16x4_f32(false, a, false, b, (short)0, c,
                                                 false, false);
}

// ---------------------------------------------------------------------------
// K1: fused theta/phi projection.  W = [w_theta;w_phi] (192x512), xf = x[b]
// (512x9216).  Each block = one (batch, 16-col strip); 3 waves handle the 3
// m-blocks of 64 rows; each wave keeps 4 fp32 16x16 accumulators.
// ---------------------------------------------------------------------------
__global__ __launch_bounds__(96)
void k1_proj(const float* __restrict__ x,
             const float* __restrict__ w_theta, const float* __restrict__ b_theta,
             const float* __restrict__ w_phi,   const float* __restrict__ b_phi,
             float* __restrict__ Bm, float* __restrict__ xr)
{
    const int wave = threadIdx.x >> 5;          // 0..2 -> m-block
    const int lane = threadIdx.x & 31;
    const int lo = lane & 15, hi = lane >> 4;
    const int bidx = blockIdx.x / NSTRIP;
    const int n0   = (blockIdx.x % NSTRIP) * 16;
    const int mbase = wave * 64;                // rows of combined W (0..191)

    const float* xb = x + (size_t)bidx * C_IN * NPIX;

    v8f acc[4] = {};
    for (int k0 = 0; k0 < C_IN; k0 += 4) {
        const int kA = k0 + 2 * hi;
        v2f bv;
        bv.x = xb[(size_t)kA * NPIX + n0 + lo];
        bv.y = xb[(size_t)(kA + 1) * NPIX + n0 + lo];
#pragma unroll
        for (int t = 0; t < 4; ++t) {
            const int m = mbase + t * 16 + lo;
            const float* wrow = (m < NNODE) ? (w_theta + (size_t)m * C_IN)
                                            : (w_phi + (size_t)(m - NNODE) * C_IN);
            v2f av; av.x = wrow[kA]; av.y = wrow[kA + 1];
            acc[t] = wmma_f32(av, bv, acc[t]);
        }
    }
#pragma unroll
    for (int t = 0; t < 4; ++t) {
#pragma unroll
        for (int r = 0; r < 8; ++r) {
            const int m = mbase + t * 16 + r + 8 * hi;
            const float v = acc[t][r];
            if (m < NNODE)
                Bm[((size_t)bidx * NNODE + m) * NPIX + n0 + lo] = v + b_theta[m];
            else
                xr[((size_t)bidx * NSTATE + (m - NNODE)) * NPIX + n0 + lo] =
                    v + b_phi[m - NNODE];
        }
    }
}

// ---------------------------------------------------------------------------
// K2: per-row sums of Bm (needed for closed-form BN mean).
// ---------------------------------------------------------------------------
__global__ __launch_bounds__(256)
void k2_rowsum(const float* __restrict__ Bm, float* __restrict__ rowsum)
{
    const int b = blockIdx.x >> 6, n = blockIdx.x & 63;
    const float* row = Bm + ((size_t)b * NNODE + n) * NPIX;
    float s = 0.f;
    for (int i = threadIdx.x; i < NPIX; i += 256) s += row[i];
    __shared__ float sm[256];
    sm[threadIdx.x] = s; __syncthreads();
    for (int off = 128; off > 0; off >>= 1) {
        if (threadIdx.x < off) sm[threadIdx.x] += sm[threadIdx.x + off];
        __syncthreads();
    }
    if (threadIdx.x == 0) rowsum[b * NNODE + n] = sm[0];
}

// ---------------------------------------------------------------------------
// K3: V = Bm.xr^T / N  (64x128) and Gram G = Bm.Bm^T (64x64), both reductions
// over K=9216.  12 n-tiles: 0..7 -> V, 8..11 -> G.  8 waves K-split + LDS add.
// ---------------------------------------------------------------------------
__global__ __launch_bounds__(256)
void k3_vg(const float* __restrict__ Bm, const float* __restrict__ xr,
           float* __restrict__ V, float* __restrict__ G)
{
    const int b   = blockIdx.x / 48;
    const int rem = blockIdx.x % 48;
    const int mt  = rem / 12;                   // m-tile 0..3
    const int nt  = rem % 12;                   // 0..7 V, 8..11 G
    const int wave = threadIdx.x >> 5, lane = threadIdx.x & 31;
    const int lo = lane & 15, hi = lane >> 4;

    const float* Arow = Bm + ((size_t)b * NNODE + mt * 16 + lo) * NPIX;
    const float* Brow;
    int n0;
    if (nt < 8) { n0 = nt * 16;       Brow = xr + ((size_t)b * NSTATE + n0 + lo) * NPIX; }
    else        { n0 = (nt - 8) * 16; Brow = Bm + ((size_t)b * NNODE  + n0 + lo) * NPIX; }

    v8f acc = {};
    const int kbeg = wave * (NPIX / 8), kend = kbeg + (NPIX / 8);
    for (int k0 = kbeg; k0 < kend; k0 += 4) {
        const int kA = k0 + 2 * hi;
        v2f av; av.x = Arow[kA]; av.y = Arow[kA + 1];
        v2f bv; bv.x = Brow[kA]; bv.y = Brow[kA + 1];
        acc = wmma_f32(av, bv, acc);
    }
    __shared__ float sm[256];
    sm[threadIdx.x] = 0.f; __syncthreads();
#pragma unroll
    for (int r = 0; r < 8; ++r) atomicAdd(&sm[(r + 8 * hi) * 16 + lo], acc[r]);
    __syncthreads();
    const int ml = threadIdx.x >> 4, nl = threadIdx.x & 15;
    const float v = sm[threadIdx.x];
    if (nt < 8) V[((size_t)b * NNODE + mt * 16 + ml) * NSTATE + n0 + nl] = v * (1.0f / (float)NPIX);
    else        G[((size_t)b * NNODE + mt * 16 + ml) * NNODE  + n0 + nl] = v;
}

// ---------------------------------------------------------------------------
// K4: graph convolution (tiny: 64x128 per batch) + Wc = w_ext . newV^T.
// Dynamic LDS: sV(64x128) | sAdj(64x64) | sSup(64x128) | sInv(64) = ~80.5 KB.
// ---------------------------------------------------------------------------
__global__ __launch_bounds__(128)
void k4_graph(const float* __restrict__ V, const float* __restrict__ gc_w,
              const float* __restrict__ gc_b, const float* __restrict__ w_ext,
              float* __restrict__ Wc)
{
    extern __shared__ float smem[];
    float* sV   = smem;                  // 64*128
    float* sAdj = sV + 64 * 128;         // 64*64
    float* sSup = sAdj + 64 * 64;        // 64*128
    float* sInv = sSup + 64 * 128;       // 64
    const int b = blockIdx.x, tid = threadIdx.x;
    const float* Vb = V + (size_t)b * NNODE * NSTATE;

    for (int i = tid; i < 64 * 128; i += 128) sV[i] = Vb[i];
    __syncthreads();
    if (tid < 64) {
        float s = 0.f;
        for (int k = 0; k < 128; ++k) { float v = sV[tid * 128 + k]; s += v * v; }
        sInv[tid] = 1.0f / fmaxf(sqrtf(s), 1e-12f);
    }
    __syncthreads();
    for (int e = tid; e < 64 * 64; e += 128) {        // adj logits (normalized)
        const int i = e >> 6, j = e & 63;
        float s = 0.f;
        for (int k = 0; k < 128; ++k) s += sV[i * 128 + k] * sV[j * 128 + k];
        sAdj[e] = s * sInv[i] * sInv[j];
    }
    __syncthreads();
    if (tid < 64) {                                    // row softmax
        float mx = -1e30f;
        for (int j = 0; j < 64; ++j) mx = fmaxf(mx, sAdj[tid * 64 + j]);
        float den = 0.f;
        for (int j = 0; j < 64; ++j) {
            float e = __expf(sAdj[tid * 64 + j] - mx);
            sAdj[tid * 64 + j] = e; den += e;
        }
        const float inv = 1.0f / den;
        for (int j = 0; j < 64; ++j) sAdj[tid * 64 + j] *= inv;
    }
    __syncthreads();
    for (int e = tid; e < 64 * 128; e += 128) {       // support = V . gc_w
        const int n = e >> 7, t = e & 127;
        float s = 0.f;
        for (int k = 0; k < 128; ++k) s += sV[n * 128 + k] * gc_w[k * 128 + t];
        sSup[e] = s;
    }
    __syncthreads();
    for (int e = tid; e < 64 * 128; e += 128) {       // newV (in-place over sV)
        const int n = e >> 7, t = e & 127;
        float s = 0.f;
        for (int k = 0; k < 64; ++k) s += sAdj[n * 64 + k] * sSup[k * 128 + t];
        sV[e] = s + gc_b[t] + sV[e];
    }
    __syncthreads();
    float* Wcb = Wc + (size_t)b * C_IN * NNODE;       // Wc = w_ext . newV^T
    for (int e = tid; e < C_IN * NNODE; e += 128) {
        const int c = e >> 6, n = e & 63;
        const float* wrow = w_ext + (size_t)c * NSTATE;
        float s = 0.f;
        for (int k = 0; k < 128; ++k) s += wrow[k] * sV[n * 128 + k];
        Wcb[e] = s;
    }
}

// ---------------------------------------------------------------------------
// K5: closed-form BN stats.  sum[c]=Σ_b Wc_b[c]·r_b ; sumsq[c]=Σ_b Wc_b[c]·G_b·Wc_b[c]^T
// ---------------------------------------------------------------------------
__global__ __launch_bounds__(256)
void k5_stats(const float* __restrict__ Wc, const float* __restrict__ G,
              const float* __restrict__ rowsum, const float* __restrict__ bn_w,
              const float* __restrict__ bn_b, float* __restrict__ scale,
              float* __restrict__ shift)
{
    const int c = blockIdx.x * 256 + threadIdx.x;
    if (c >= C_IN) return;
    float sum = 0.f, sumsq = 0.f;
    for (int b = 0; b < BSZ; ++b) {
        const float* w  = Wc + ((size_t)b * C_IN + c) * NNODE;
        const float* Gb = G + (size_t)b * NNODE * NNODE;
        const float* rs = rowsum + b * NNODE;
        float wl[64];
#pragma unroll
        for (int i = 0; i < 64; ++i) wl[i] = w[i];
        for (int i = 0; i < 64; ++i) {
            float ti = 0.f;
            for (int j = 0; j < 64; ++j) ti += Gb[i * 64 + j] * wl[j];
            sumsq += wl[i] * ti;
            sum   += wl[i] * rs[i];
        }
    }
    const float invcnt = 1.0f / ((float)BSZ * (float)NPIX);
    const float mean = sum * invcnt;
    const float var  = sumsq * invcnt - mean * mean;
    const float sc   = bn_w[c] * rsqrtf(var + 1e-5f);
    scale[c] = sc;
    shift[c] = bn_b[c] - mean * sc;
}

// ---------------------------------------------------------------------------
// K6: out = (Wc_b . Bm_b) * scale + shift + x   (Yc never materialized).
// M=512, K=64, N=9216.  4 waves/block, wave = one 64-row m-band.
// ---------------------------------------------------------------------------
__global__ __launch_bounds__(128)
void k6_out(const float* __restrict__ Wc, const float* __restrict__ Bm,
            const float* __restrict__ x, const float* __restrict__ scale,
            const float* __restrict__ shift, float* __restrict__ out)
{
    const int b   = blockIdx.x / (NSTRIP * 2);
    const int rem = blockIdx.x % (NSTRIP * 2);
    const int n0  = (rem >> 1) * 16;
    const int wave = threadIdx.x >> 5, lane = threadIdx.x & 31;
    const int lo = lane & 15, hi = lane >> 4;
    const int mband = (rem & 1) * 4 + wave;           // 0..7 -> rows mband*64

    const float* Wcb = Wc + (size_t)b * C_IN * NNODE;
    const float* Bmb = Bm + (size_t)b * NNODE * NPIX;

    v8f acc[4] = {};
    for (int k0 = 0; k0 < NNODE; k0 += 4) {
        const int kA = k0 + 2 * hi;
        v2f bv;
        bv.x = Bmb[(size_t)kA * NPIX + n0 + lo];
        bv.y = Bmb[(size_t)(kA + 1) * NPIX + n0 + lo];
#pragma unroll
        for (int t = 0; t < 4; ++t) {
            const int m = mband * 64 + t * 16 + lo;
            v2f av; av.x = Wcb[m * NNODE + kA]; av.y = Wcb[m * NNODE + kA + 1];
            acc[t] = wmma_f32(av, bv, acc[t]);
        }
    }
    const float* xb = x + (size_t)b * C_IN * NPIX;
    float* ob = out + (size_t)b * C_IN * NPIX;
#pragma unroll
    for (int t = 0; t < 4; ++t) {
#pragma unroll
        for (int r = 0; r < 8; ++r) {
            const int c = mband * 64 + t * 16 + r + 8 * hi;
            const size_t idx = (size_t)c * NPIX + n0 + lo;
            ob[idx] = acc[t][r] * scale[c] + shift[c] + xb[idx];
        }
    }
}

// ---------------------------------------------------------------------------
extern "C" void kernel_launch(void* const* d_in, const int* in_sizes, int n_in,
                              void* d_out, int out_size, void* d_ws, size_t ws_size,
                              hipStream_t stream)
{
    const float* x       = (const float*)d_in[0];
    const float* w_theta = (const float*)d_in[1];
    const float* b_theta = (const float*)d_in[2];
    const float* w_phi   = (const float*)d_in[3];
    const float* b_phi   = (const float*)d_in[4];
    const float* gc_w    = (const float*)d_in[5];
    const float* gc_b    = (const float*)d_in[6];
    const float* w_ext   = (const float*)d_in[7];
    const float* bn_w    = (const float*)d_in[8];
    const float* bn_b    = (const float*)d_in[9];
    float* out = (float*)d_out;

    float* ws = (float*)d_ws;
    size_t off = 0;
    float* Bm     = ws + off; off += (size_t)BSZ * NNODE  * NPIX;   // 18.9 MB
    float* xr     = ws + off; off += (size_t)BSZ * NSTATE * NPIX;   // 37.7 MB
    float* V      = ws + off; off += (size_t)BSZ * NNODE * NSTATE;
    float* G      = ws + off; off += (size_t)BSZ * NNODE * NNODE;
    float* rowsum = ws + off; off += (size_t)BSZ * NNODE;
    float* Wc     = ws + off; off += (size_t)BSZ * C_IN * NNODE;    // 1 MB
    float* scale  = ws + off; off += C_IN;
    float* shift  = ws + off; off += C_IN;

    k1_proj  <<<BSZ * NSTRIP, 96, 0, stream>>>(x, w_theta, b_theta, w_phi, b_phi, Bm, xr);
    k2_rowsum<<<BSZ * NNODE, 256, 0, stream>>>(Bm, rowsum);
    k3_vg    <<<BSZ * 48, 256, 0, stream>>>(Bm, xr, V, G);
    k4_graph <<<BSZ, 128, (64 * 128 + 64 * 64 + 64 * 128 + 64) * sizeof(float), stream>>>(
        V, gc_w, gc_b, w_ext, Wc);
    k5_stats <<<2, 256, 0, stream>>>(Wc, G, rowsum, bn_w, bn_b, scale, shift);
    k6_out   <<<BSZ * NSTRIP * 2, 128, 0, stream>>>(Wc, Bm, x, scale, shift, out);
}